// Fractal1DPreprocessor_4509715661423
// MI455X (gfx1250) — compile-verified
//
#include <hip/hip_runtime.h>
#include <math.h>

// Problem constants (from the reference "generic" preset)
static constexpr int   LROW   = 16384;   // row length
static constexpr int   NT     = 512;     // threads per block (16 wave32 waves)
static constexpr int   CHUNK  = LROW / NT;        // 32 elements per thread
static constexpr int   VCHUNK = LROW / (NT * 4);  // 8 float4 per thread
static constexpr float TRENDC = 0.55f;
static constexpr float DETC   = 0.88f;
static constexpr float THRESH = 3.5f;
static constexpr float DAMP   = 0.4f;
static constexpr float EPS    = 1e-6f;

// Types matching the async-load builtin signature revealed by the compiler:
// parameter is int4 (__vector_size__) pointers in AS1 (global) / AS3 (LDS).
typedef int v4i __attribute__((__vector_size__(4 * sizeof(int))));
typedef __attribute__((address_space(1))) v4i* g_v4i_ptr;
typedef __attribute__((address_space(3))) v4i* l_v4i_ptr;

__device__ __forceinline__ int reflect_idx(int i) {
    // jnp.pad mode="reflect": -1 -> 1, -2 -> 2, L -> L-2, L+1 -> L-3
    i = (i < 0) ? -i : i;
    i = (i >= LROW) ? (2 * LROW - 2 - i) : i;
    return i;
}

// Fused block reduction of three quantities (wave32 shuffle + LDS scratch).
__device__ __forceinline__ void block_reduce3(float& a, float& b, float& c,
                                              float* scratch) {
    #pragma unroll
    for (int off = 16; off > 0; off >>= 1) {
        a += __shfl_down(a, off, 32);
        b += __shfl_down(b, off, 32);
        c += __shfl_down(c, off, 32);
    }
    const int tid  = threadIdx.x;
    const int wave = tid >> 5;
    if ((tid & 31) == 0) {
        scratch[wave]      = a;
        scratch[16 + wave] = b;
        scratch[32 + wave] = c;
    }
    __syncthreads();
    if (tid == 0) {
        float ra = 0.f, rb = 0.f, rc = 0.f;
        #pragma unroll
        for (int w = 0; w < NT / 32; ++w) {
            ra += scratch[w];
            rb += scratch[16 + w];
            rc += scratch[32 + w];
        }
        scratch[48] = ra; scratch[49] = rb; scratch[50] = rc;
    }
    __syncthreads();
    a = scratch[48]; b = scratch[49]; c = scratch[50];
    __syncthreads();   // protect scratch reuse (WAR) for the next reduction
}

__global__ __launch_bounds__(NT)
void fractal1d_kernel(const float* __restrict__ x, float* __restrict__ out) {
    __shared__ float row[LROW];     // 64 KB row buffer (LDS; 320 KB/WGP on CDNA5)
    __shared__ float scratch[64];

    const int t = threadIdx.x;
    const long long rbase = (long long)blockIdx.x * (long long)LROW;
    const float* __restrict__ src = x + rbase;
    float* __restrict__ dst = out + rbase;

    // ---- Stage 1: bulk copy row -> LDS via CDNA5 async global->LDS path ----
#if defined(__gfx1250__) && __has_builtin(__builtin_amdgcn_global_load_async_to_lds_b128)
    {
        float* srcnc = const_cast<float*>(src);   // builtin param is non-const
        #pragma unroll
        for (int c = 0; c < VCHUNK; ++c) {
            const int off = (c * NT + t) * 4;     // float index, 16B aligned
            __builtin_amdgcn_global_load_async_to_lds_b128(
                (g_v4i_ptr)(srcnc + off),         // addrspacecast generic->AS1
                (l_v4i_ptr)(row + off),           // addrspacecast generic->AS3
                /*offset=*/0, /*cpol=*/0);
        }
    }
  #if __has_builtin(__builtin_amdgcn_s_wait_asynccnt)
    __builtin_amdgcn_s_wait_asynccnt(0);
  #else
    asm volatile("s_wait_asynccnt 0" ::: "memory");
  #endif
#else
    #pragma unroll
    for (int c = 0; c < VCHUNK; ++c) {
        const int off = (c * NT + t) * 4;
        *(float4*)(row + off) = *(const float4*)(src + off);
    }
#endif
    __syncthreads();

    const float invN = 1.0f / (float)LROW;
    float dreg[CHUNK];   // per-thread detail values (registers)

    for (int iter = 0; iter < 2; ++iter) {
        // ---- Pass A: stencil + stats (row is read-only here) ----
        float sx = 0.f, sd = 0.f, sd2 = 0.f;
        #pragma unroll
        for (int j = 0; j < CHUNK; ++j) {
            const int i = j * NT + t;      // strided: bank-conflict-free
            const float xm2 = row[reflect_idx(i - 2)];
            const float xm1 = row[reflect_idx(i - 1)];
            const float x0  = row[i];
            const float xp1 = row[reflect_idx(i + 1)];
            const float xp2 = row[reflect_idx(i + 2)];
            const float trend = (xm2 + xm1 + x0 + xp1 + xp2) * 0.2f;
            const float sm = TRENDC * trend + (1.0f - TRENDC) * x0;
            const float d = x0 - sm;
            dreg[j] = d;
            sx += x0; sd += d; sd2 += d * d;
        }
        block_reduce3(sx, sd, sd2, scratch);   // contains barriers: A/B hazard safe

        const float md   = sd * invN;
        const float var  = (sd2 - sd * md) / (float)(LROW - 1); // unbiased (ddof=1)
        const float stdd = fmaxf(sqrtf(fmaxf(var, 0.f)), EPS);
        const float thr  = THRESH * stdd;
        // fold the initial mean-centering into iteration 0's write-back:
        // detail/std are shift-invariant, only smoothed carries the -mean.
        const float msub = (iter == 0) ? sx * invN : 0.f;

        // ---- Pass B: damp + blend; own elements only (no cross-thread hazard) ----
        float sp = 0.f, sp2 = 0.f, dzero = 0.f;
        #pragma unroll
        for (int j = 0; j < CHUNK; ++j) {
            const int i = j * NT + t;
            const float x0 = row[i];
            const float d  = dreg[j];
            const float sm = x0 - d;                     // smoothed, recovered free
            const float dd = (fabsf(d) > thr) ? d * DAMP : d;
            const float nv = (sm - msub) + DETC * dd;
            row[i] = nv;
            sp += nv; sp2 += nv * nv;
        }
        __syncthreads();

        if (iter == 1) {
            // ---- Final normalization stats came for free from pass B ----
            block_reduce3(sp, sp2, dzero, scratch);
            const float m    = sp * invN;
            const float var2 = (sp2 - sp * m) / (float)(LROW - 1);
            const float s    = fmaxf(sqrtf(fmaxf(var2, 0.f)), EPS);
            const float invs = 1.0f / s;
            // ---- Stage 3: coalesced b128 stores straight from LDS ----
            #pragma unroll
            for (int c = 0; c < VCHUNK; ++c) {
                const int off = (c * NT + t) * 4;
                const float4 v = *(const float4*)(row + off);
                float4 o;
                o.x = (v.x - m) * invs;
                o.y = (v.y - m) * invs;
                o.z = (v.z - m) * invs;
                o.w = (v.w - m) * invs;
                *(float4*)(dst + off) = o;
            }
        }
    }
}

extern "C" void kernel_launch(void* const* d_in, const int* in_sizes, int n_in,
                              void* d_out, int out_size, void* d_ws, size_t ws_size,
                              hipStream_t stream) {
    (void)n_in; (void)out_size; (void)d_ws; (void)ws_size;
    const float* x = (const float*)d_in[0];
    float* out = (float*)d_out;
    const int rows = in_sizes[0] / LROW;   // 4096 for the reference shape
    fractal1d_kernel<<<rows, NT, 0, stream>>>(x, out);
}